// SecondOrderFeatureInteraction_66236985639380
// MI455X (gfx1250) — compile-verified
//
#include <hip/hip_runtime.h>
#include <math.h>

// SecondOrderFeatureInteraction: per-sample gram matrix G = A·A^T (A: 27x128 fp32),
// output strictly-upper-triangular entries, p = j*(j-1)/2 + i  (i < j).
//
// Memory-bound: ~250 MB total traffic -> ~10.7 us at 23.3 TB/s. Compute via
// V_WMMA_F32_16X16X4_F32 (fp32-exact, 3 accumulator tiles of the padded 32x32 gram).

typedef __attribute__((ext_vector_type(2))) float v2f;
typedef __attribute__((ext_vector_type(8))) float v8f;

#define FEAS   27
#define EDIM   128
#define NPAIR  351          // 27*26/2
#define WAVES_PER_BLOCK 4
#define GSTRIDE 28          // LDS row stride (floats) to avoid bank conflicts

__global__ __launch_bounds__(32 * WAVES_PER_BLOCK)
void gram_upper_tri_kernel(const float* __restrict__ in,
                           float* __restrict__ out,
                           int batch) {
    __shared__ float gsh[WAVES_PER_BLOCK][FEAS * GSTRIDE];

    const int lane = threadIdx.x & 31;
    const int wave = threadIdx.x >> 5;
    const int b    = blockIdx.x * WAVES_PER_BLOCK + wave;
    if (b >= batch) return;            // wave-uniform: EXEC stays all-1s for WMMA

    const int m    = lane & 15;        // row/col within 16-tile
    const int half = lane >> 4;        // selects K-pair {0,1} vs {2,3}

    const float* base = in + (size_t)b * (FEAS * EDIM);

    // fp32 WMMA fragment layout (16x4 A == 4x16 B for symmetric product):
    //   lane l: row/col = l%16, two consecutive K elements at offset k + 2*(l/16)
    const float* p0 = base + m * EDIM + 2 * half;                 // rows 0..15
    int r1 = 16 + m; if (r1 > FEAS - 1) r1 = FEAS - 1;            // clamp rows 27..31
    const float* p1 = base + r1 * EDIM + 2 * half;                // rows 16..26 (+clamped)

    v8f acc00 = {};   // G[0:16, 0:16]
    v8f acc01 = {};   // G[0:16, 16:32]
    v8f acc11 = {};   // G[16:32, 16:32]

#pragma unroll
    for (int k = 0; k < EDIM; k += 4) {
        v2f f0 = *(const v2f*)(p0 + k);
        v2f f1 = *(const v2f*)(p1 + k);
        acc00 = __builtin_amdgcn_wmma_f32_16x16x4_f32(false, f0, false, f0,
                                                      (short)0, acc00, false, false);
        acc01 = __builtin_amdgcn_wmma_f32_16x16x4_f32(false, f0, false, f1,
                                                      (short)0, acc01, false, false);
        acc11 = __builtin_amdgcn_wmma_f32_16x16x4_f32(false, f1, false, f1,
                                                      (short)0, acc11, false, false);
    }

    // C/D layout: VGPR r, lane l -> row = r + 8*(l/16), col = l%16 (per 16x16 tile)
    float* g = &gsh[wave][0];
    const int row0 = 8 * half;
#pragma unroll
    for (int r = 0; r < 8; ++r) {
        g[(row0 + r) * GSTRIDE + m] = acc00[r];
        if (16 + m < FEAS)
            g[(row0 + r) * GSTRIDE + 16 + m] = acc01[r];
        if ((16 + row0 + r < FEAS) && (16 + m < FEAS))
            g[(16 + row0 + r) * GSTRIDE + 16 + m] = acc11[r];
    }
    __syncthreads();

    // Emit strictly-upper-triangular entries: p = j*(j-1)/2 + i, i < j
    float* dst = out + (size_t)b * NPAIR;
    for (int p = lane; p < NPAIR; p += 32) {
        int j = (int)((1.0f + sqrtf(8.0f * (float)p + 1.0f)) * 0.5f);
        while (j * (j - 1) / 2 > p)  --j;   // integer-exact fixup of float estimate
        while (j * (j + 1) / 2 <= p) ++j;
        int i = p - j * (j - 1) / 2;
        dst[p] = g[i * GSTRIDE + j];
    }
}

extern "C" void kernel_launch(void* const* d_in, const int* in_sizes, int n_in,
                              void* d_out, int out_size, void* d_ws, size_t ws_size,
                              hipStream_t stream) {
    const float* in = (const float*)d_in[0];
    // d_in[1] is num_feas (scalar, == 27) — fixed at compile time.
    float* out = (float*)d_out;
    int batch = in_sizes[0] / (FEAS * EDIM);   // 16384
    int blocks = (batch + WAVES_PER_BLOCK - 1) / WAVES_PER_BLOCK;
    gram_upper_tri_kernel<<<blocks, 32 * WAVES_PER_BLOCK, 0, stream>>>(in, out, batch);
}